// LatentCoulombLongRange_66881230733959
// MI455X (gfx1250) — compile-verified
//
#include <hip/hip_runtime.h>

typedef __attribute__((ext_vector_type(2))) float v2f;
typedef __attribute__((ext_vector_type(8))) float v8f;

#define N_ATOMS 8192
#define FEAT    128
#define HID     64
#define NGRAPH  8
#define JSPLIT  8
#define LOG2E   1.44269504088896340736f

// ---------------------------------------------------------------------------
// init: zero output and per-graph accumulators
// ---------------------------------------------------------------------------
__global__ void lclr_init_kernel(float* __restrict__ out,
                                 float* __restrict__ sumq,
                                 float* __restrict__ cntg) {
    int i = blockIdx.x * blockDim.x + threadIdx.x;
    if (i < N_ATOMS) out[i] = 0.0f;
    if (i < NGRAPH) { sumq[i] = 0.0f; cntg[i] = 0.0f; }
}

// ---------------------------------------------------------------------------
// count atoms per graph
// ---------------------------------------------------------------------------
__global__ void lclr_count_kernel(const int* __restrict__ batch,
                                  float* __restrict__ cntg) {
    int i = blockIdx.x * blockDim.x + threadIdx.x;
    if (i < N_ATOMS) atomicAdd(&cntg[batch[i]], 1.0f);
}

// ---------------------------------------------------------------------------
// charge head: q = silu(x@W1 + b1) @ W2 + b2, fused in-register.
// One wave per 16-row tile of x. Uses V_WMMA_F32_16X16X4_F32 (full f32).
//
// A (16x4 f32) layout: lanes 0-15 & 16-31 both hold rows M=0..15 (M=lane&15);
//   VGPR v, lane-half g: K = g*2 + v.
// B (4x16 f32) layout (mirrored): VGPR v, lane-half g: K = g*2 + v, N = lane&15.
// C/D (16x16 f32): VGPR r: lanes 0-15 -> M=r, lanes 16-31 -> M=r+8; N = lane&15.
// ---------------------------------------------------------------------------
__global__ void lclr_charge_kernel(const float* __restrict__ x,
                                   const float* __restrict__ W1,
                                   const float* __restrict__ b1,
                                   const float* __restrict__ W2,
                                   const float* __restrict__ b2,
                                   const int*   __restrict__ batch,
                                   float* __restrict__ q,
                                   float* __restrict__ sumq) {
    const int lane  = threadIdx.x & 31;
    const int wave  = threadIdx.x >> 5;
    const int tile  = blockIdx.x * 8 + wave;     // 512 tiles of 16 rows
    const int m0    = tile * 16;
    const int col   = lane & 15;
    const int khalf = lane >> 4;                 // 0: lanes 0-15, 1: lanes 16-31

    v8f acc[4] = {};                             // 4 column tiles: cols t*16+col

    const float* xrow = x + (size_t)(m0 + col) * FEAT;
    for (int kb = 0; kb < FEAT; kb += 4) {
        v2f a;
        a.x = xrow[kb + khalf * 2 + 0];
        a.y = xrow[kb + khalf * 2 + 1];
#pragma unroll
        for (int t = 0; t < 4; ++t) {
            v2f bm;
            bm.x = W1[(size_t)(kb + khalf * 2 + 0) * HID + t * 16 + col];
            bm.y = W1[(size_t)(kb + khalf * 2 + 1) * HID + t * 16 + col];
            acc[t] = __builtin_amdgcn_wmma_f32_16x16x4_f32(
                false, a, false, bm, (short)0, acc[t], false, false);
        }
    }

    // epilogue: bias + SiLU + dot with W2 column values
    float b1c[4], w2c[4];
#pragma unroll
    for (int t = 0; t < 4; ++t) {
        b1c[t] = b1[t * 16 + col];
        w2c[t] = W2[t * 16 + col];
    }
    float partial[8];
#pragma unroll
    for (int r = 0; r < 8; ++r) partial[r] = 0.0f;
#pragma unroll
    for (int t = 0; t < 4; ++t) {
#pragma unroll
        for (int r = 0; r < 8; ++r) {
            float h = acc[t][r] + b1c[t];
            float sig = __builtin_amdgcn_rcpf(1.0f + __builtin_amdgcn_exp2f(-LOG2E * h));
            partial[r] += (h * sig) * w2c[t];
        }
    }
    // reduce over the 16 lanes of each half-wave (xor masks < 16 stay in-half)
#pragma unroll
    for (int off = 1; off < 16; off <<= 1) {
#pragma unroll
        for (int r = 0; r < 8; ++r)
            partial[r] += __shfl_xor(partial[r], off, 32);
    }
    const float bias2 = b2[0];
    if (lane == 0) {
#pragma unroll
        for (int r = 0; r < 8; ++r) {
            float qv = partial[r] + bias2;
            q[m0 + r] = qv;
            atomicAdd(&sumq[batch[m0 + r]], qv);
        }
    } else if (lane == 16) {
#pragma unroll
        for (int r = 0; r < 8; ++r) {
            float qv = partial[r] + bias2;
            q[m0 + 8 + r] = qv;
            atomicAdd(&sumq[batch[m0 + 8 + r]], qv);
        }
    }
}

// ---------------------------------------------------------------------------
// neutralize: q_adj = q - mean_q[graph]
// ---------------------------------------------------------------------------
__global__ void lclr_adjust_kernel(const float* __restrict__ q,
                                   const int*   __restrict__ batch,
                                   const float* __restrict__ sumq,
                                   const float* __restrict__ cntg,
                                   float* __restrict__ qadj) {
    int i = blockIdx.x * blockDim.x + threadIdx.x;
    if (i >= N_ATOMS) return;
    int b = batch[i];
    qadj[i] = q[i] - sumq[b] / fmaxf(cntg[b], 1.0f);
}

// ---------------------------------------------------------------------------
// pairwise screened Coulomb. Thread owns atom i; j streamed through LDS in
// 256-wide tiles. batch is sorted, so a tile whose [batch[first],batch[last]]
// range excludes b_i is skipped entirely (recovers the ~8x segment sparsity
// without data-dependent bounds). j-range split over blockIdx.y for occupancy;
// partial sums combined with global_atomic_add_f32.
// ---------------------------------------------------------------------------
__global__ void lclr_pair_kernel(const float* __restrict__ pos,
                                 const float* __restrict__ qadj,
                                 const int*   __restrict__ batch,
                                 const float* __restrict__ screening_p,
                                 const float* __restrict__ softening_p,
                                 float* __restrict__ out) {
    __shared__ float sx[256], sy[256], sz[256], sq[256];
    __shared__ int   sb[256];

    const int i  = blockIdx.x * 256 + threadIdx.x;
    const int bi = batch[i];
    const float px = pos[i * 3 + 0];
    const float py = pos[i * 3 + 1];
    const float pz = pos[i * 3 + 2];
    const float qi = qadj[i];

    const float soft  = softening_p[0];
    const float soft2 = soft * soft;
    const float kexp  = -screening_p[0] * LOG2E;   // exp(-s*r) = exp2(kexp*r)

    const int clo = blockIdx.y * (N_ATOMS / JSPLIT);
    const int chi = clo + (N_ATOMS / JSPLIT);

    float accum = 0.0f;
    for (int tl = clo; tl < chi; tl += 256) {
        const int j = tl + threadIdx.x;
        sx[threadIdx.x] = pos[j * 3 + 0];
        sy[threadIdx.x] = pos[j * 3 + 1];
        sz[threadIdx.x] = pos[j * 3 + 2];
        sq[threadIdx.x] = qadj[j];
        sb[threadIdx.x] = batch[j];
        __syncthreads();

        const int bmin = sb[0], bmax = sb[255];    // sorted batch -> tile range
        if (bi >= bmin && bi <= bmax) {
#pragma unroll 4
            for (int t = 0; t < 256; ++t) {
                float dx = sx[t] - px;
                float dy = sy[t] - py;
                float dz = sz[t] - pz;
                float r2 = fmaf(dx, dx, fmaf(dy, dy, fmaf(dz, dz, soft2)));
                float inv_r = __builtin_amdgcn_rsqf(r2);   // v_rsq_f32
                float r     = r2 * inv_r;
                float kern  = __builtin_amdgcn_exp2f(kexp * r) * inv_r;
                bool valid  = (sb[t] == bi) && ((tl + t) != i);
                accum += valid ? sq[t] * kern : 0.0f;
            }
        }
        __syncthreads();
    }
    atomicAdd(&out[i], 0.5f * qi * accum);
}

// ---------------------------------------------------------------------------
// host launch
// ---------------------------------------------------------------------------
extern "C" void kernel_launch(void* const* d_in, const int* in_sizes, int n_in,
                              void* d_out, int out_size, void* d_ws, size_t ws_size,
                              hipStream_t stream) {
    const float* x    = (const float*)d_in[0];
    const float* pos  = (const float*)d_in[1];
    // d_in[2] = cell (unused: nonperiodic kernel)
    const float* W1   = (const float*)d_in[3];
    const float* b1   = (const float*)d_in[4];
    const float* W2   = (const float*)d_in[5];
    const float* b2   = (const float*)d_in[6];
    const float* scr  = (const float*)d_in[7];
    const float* soft = (const float*)d_in[8];
    const int*   bat  = (const int*)d_in[9];
    float* out = (float*)d_out;

    float* q    = (float*)d_ws;          // N
    float* qadj = q + N_ATOMS;           // N
    float* sumq = qadj + N_ATOMS;        // NGRAPH
    float* cntg = sumq + NGRAPH;         // NGRAPH

    lclr_init_kernel<<<(N_ATOMS + 255) / 256, 256, 0, stream>>>(out, sumq, cntg);
    lclr_count_kernel<<<N_ATOMS / 256, 256, 0, stream>>>(bat, cntg);
    lclr_charge_kernel<<<N_ATOMS / 16 / 8, 256, 0, stream>>>(x, W1, b1, W2, b2,
                                                             bat, q, sumq);
    lclr_adjust_kernel<<<N_ATOMS / 256, 256, 0, stream>>>(q, bat, sumq, cntg, qadj);
    dim3 grid(N_ATOMS / 256, JSPLIT);
    lclr_pair_kernel<<<grid, 256, 0, stream>>>(pos, qadj, bat, scr, soft, out);
}